// Attention_68934225101259
// MI455X (gfx1250) — compile-verified
//
#include <hip/hip_runtime.h>
#include <cmath>

// ---- problem sizes (fixed by the reference) ----
#define RNN       1024
#define ATT_HID   512
#define ATT_FEAT  2048
#define ENC       1024
#define BATCH     256
#define ATT_SIZE  196
#define WPAD      256   // padded per-row stride for softmax weights in ws

typedef __attribute__((ext_vector_type(2))) float v2f;
typedef __attribute__((ext_vector_type(4))) float v4f;
typedef __attribute__((ext_vector_type(8))) float v8f;

__device__ __forceinline__ float sigmoidf_(float x) {
    return 1.0f / (1.0f + __expf(-x));
}

// =====================================================================
// Kernel 1: att_h = h @ W_h2att + b_h2att   (M=256, K=1024, N=1024)
// f32 WMMA 16x16x4, one wave per 16x16 output tile, 4 waves/block.
// A frag: lane<16 -> M=l, K={k,k+1}; lane>=16 -> K={k+2,k+3}
// B frag: same K mapping, N = l; C/D: vgpr r -> row r (lanes 0-15) / r+8
// =====================================================================
__global__ __launch_bounds__(128) void k_gemm_h2att(
    const float* __restrict__ A,     // [256, 1024]
    const float* __restrict__ B,     // [1024, 1024] row-major (K,N)
    const float* __restrict__ bias,  // [1024]
    float* __restrict__ C)           // [256, 1024]
{
    const int K = RNN, N = 2 * ATT_HID;
    const int lane = threadIdx.x & 31;
    const int wave = threadIdx.x >> 5;
    const int n0 = (blockIdx.x * 4 + wave) * 16;
    const int m0 = blockIdx.y * 16;
    const int hh = lane >> 4;   // which K half-pair this lane owns
    const int l  = lane & 15;

    const float* arow = A + (size_t)(m0 + l) * K;
    v8f acc = {};
#pragma unroll 8
    for (int k = 0; k < K; k += 4) {
        const int kk = k + 2 * hh;
        v2f a; a.x = arow[kk];
               a.y = arow[kk + 1];
        v2f b; b.x = B[(size_t)kk       * N + n0 + l];
               b.y = B[(size_t)(kk + 1) * N + n0 + l];
        acc = __builtin_amdgcn_wmma_f32_16x16x4_f32(
                  false, a, false, b, (short)0, acc, false, false);
    }
    const float bn = bias[n0 + l];
#pragma unroll
    for (int r = 0; r < 8; ++r)
        C[(size_t)(m0 + r + 8 * hh) * N + n0 + l] = acc[r] + bn;
}

// =====================================================================
// Kernel 2: scores[b,s] = sum_j tanh(p+ah)*sigmoid(p'+ah')*w_alpha + b_alpha
//           then softmax over s.  One workgroup per batch row b.
// p_att_feats is streamed once -> non-temporal loads.
// =====================================================================
__global__ __launch_bounds__(256) void k_scores_softmax(
    const float* __restrict__ p,        // [B, S, 1024]
    const float* __restrict__ att_h,    // [B, 1024]
    const float* __restrict__ w_alpha,  // [512]
    const float* __restrict__ b_alpha,  // [1]
    float* __restrict__ weight)         // [B, WPAD]
{
    __shared__ float ah[2 * ATT_HID];
    __shared__ float wa[ATT_HID];
    __shared__ float sc[WPAD];
    __shared__ float red[256];

    const int b   = blockIdx.x;
    const int tid = threadIdx.x;

    const float* ahg = att_h + (size_t)b * (2 * ATT_HID);
    for (int i = tid; i < 2 * ATT_HID; i += 256) ah[i] = ahg[i];
    for (int i = tid; i < ATT_HID;     i += 256) wa[i] = w_alpha[i];
    __syncthreads();

    const int lane = tid & 31, wv = tid >> 5;
    const float balpha = b_alpha[0];

    for (int s = wv; s < ATT_SIZE; s += 8) {
        const float* prow = p + ((size_t)b * ATT_SIZE + s) * (2 * ATT_HID);
        float acc = 0.0f;
        for (int j = lane; j < ATT_HID; j += 32) {
            const float d0 = __builtin_nontemporal_load(prow + j)            + ah[j];
            const float d1 = __builtin_nontemporal_load(prow + ATT_HID + j)  + ah[ATT_HID + j];
            acc = fmaf(tanhf(d0) * sigmoidf_(d1), wa[j], acc);
        }
#pragma unroll
        for (int o = 16; o > 0; o >>= 1) acc += __shfl_xor(acc, o, 32);
        if (lane == 0) sc[s] = acc + balpha;
    }
    __syncthreads();

    // block softmax over 196 entries
    const float v = (tid < ATT_SIZE) ? sc[tid] : -3.402823466e38f;
    red[tid] = v; __syncthreads();
    for (int off = 128; off > 0; off >>= 1) {
        if (tid < off) red[tid] = fmaxf(red[tid], red[tid + off]);
        __syncthreads();
    }
    const float m = red[0]; __syncthreads();
    const float e = (tid < ATT_SIZE) ? __expf(v - m) : 0.0f;
    red[tid] = e; __syncthreads();
    for (int off = 128; off > 0; off >>= 1) {
        if (tid < off) red[tid] += red[tid + off];
        __syncthreads();
    }
    const float sum = red[0];
    weight[(size_t)b * WPAD + tid] = (tid < ATT_SIZE) ? (e / sum) : 0.0f;
}

// =====================================================================
// Kernel 3: att_res[b,f] = sum_s weight[b,s] * att_feats[b,s,f]
// The 411 MB stream that sets total runtime: NT float4 loads + prefetch.
// One workgroup per b; each thread owns two float4 column slices.
// =====================================================================
__global__ __launch_bounds__(256) void k_att_res(
    const float* __restrict__ feats,   // [B, S, 2048]
    const float* __restrict__ weight,  // [B, WPAD]
    float* __restrict__ att_res)       // [B, 2048]
{
    __shared__ float wsm[ATT_SIZE];
    const int b = blockIdx.x, tid = threadIdx.x;
    if (tid < ATT_SIZE) wsm[tid] = weight[(size_t)b * WPAD + tid];
    __syncthreads();

    const float* base = feats + (size_t)b * ATT_SIZE * ATT_FEAT;
    v4f a0 = {}; v4f a1 = {};
    for (int s = 0; s < ATT_SIZE; ++s) {
        const v4f* row = (const v4f*)(base + (size_t)s * ATT_FEAT);
        if (s + 4 < ATT_SIZE)
            __builtin_prefetch(base + (size_t)(s + 4) * ATT_FEAT + tid * 4, 0, 0);
        const float w = wsm[s];
        const v4f f0 = __builtin_nontemporal_load(row + tid);
        const v4f f1 = __builtin_nontemporal_load(row + tid + 256);
        a0.x = fmaf(w, f0.x, a0.x); a0.y = fmaf(w, f0.y, a0.y);
        a0.z = fmaf(w, f0.z, a0.z); a0.w = fmaf(w, f0.w, a0.w);
        a1.x = fmaf(w, f1.x, a1.x); a1.y = fmaf(w, f1.y, a1.y);
        a1.z = fmaf(w, f1.z, a1.z); a1.w = fmaf(w, f1.w, a1.w);
    }
    v4f* out = (v4f*)(att_res + (size_t)b * ATT_FEAT);
    out[tid]       = a0;
    out[tid + 256] = a1;
}

// =====================================================================
// Kernel 4: out = att_res @ W_out + b_out ; y = tanh(out[:, :1024]) *
//           sigmoid(out[:, 1024:]).  Each wave accumulates TWO WMMA tiles
//           (cols n0 and n0+1024) so the gate is fused — no HBM roundtrip.
// =====================================================================
__global__ __launch_bounds__(128) void k_gemm_out_gate(
    const float* __restrict__ A,     // att_res [256, 2048]
    const float* __restrict__ B,     // W_out   [2048, 2048] (K,N)
    const float* __restrict__ bias,  // [2048]
    float* __restrict__ Y)           // [256, 1024]
{
    const int K = ATT_FEAT, N = 2 * ENC;
    const int lane = threadIdx.x & 31;
    const int wave = threadIdx.x >> 5;
    const int n0 = (blockIdx.x * 4 + wave) * 16;   // 0..1008 (tanh half)
    const int m0 = blockIdx.y * 16;
    const int hh = lane >> 4;
    const int l  = lane & 15;

    const float* arow = A + (size_t)(m0 + l) * K;
    v8f acc0 = {}; v8f acc1 = {};
#pragma unroll 4
    for (int k = 0; k < K; k += 4) {
        const int kk = k + 2 * hh;
        v2f a;  a.x  = arow[kk];
                a.y  = arow[kk + 1];
        v2f b0; b0.x = B[(size_t)kk       * N + n0 + l];
                b0.y = B[(size_t)(kk + 1) * N + n0 + l];
        v2f b1; b1.x = B[(size_t)kk       * N + ENC + n0 + l];
                b1.y = B[(size_t)(kk + 1) * N + ENC + n0 + l];
        acc0 = __builtin_amdgcn_wmma_f32_16x16x4_f32(
                   false, a, false, b0, (short)0, acc0, false, false);
        acc1 = __builtin_amdgcn_wmma_f32_16x16x4_f32(
                   false, a, false, b1, (short)0, acc1, false, false);
    }
    const float bt = bias[n0 + l];
    const float bs = bias[ENC + n0 + l];
#pragma unroll
    for (int r = 0; r < 8; ++r) {
        const float t = tanhf(acc0[r] + bt);
        const float g = sigmoidf_(acc1[r] + bs);
        Y[(size_t)(m0 + r + 8 * hh) * RNN + n0 + l] = t * g;
    }
}

// =====================================================================
extern "C" void kernel_launch(void* const* d_in, const int* in_sizes, int n_in,
                              void* d_out, int out_size, void* d_ws, size_t ws_size,
                              hipStream_t stream)
{
    const float* h           = (const float*)d_in[0];
    const float* att_feats   = (const float*)d_in[1];
    const float* p_att_feats = (const float*)d_in[2];
    const float* W_h2att     = (const float*)d_in[3];
    const float* b_h2att     = (const float*)d_in[4];
    const float* w_alpha     = (const float*)d_in[5];
    const float* b_alpha     = (const float*)d_in[6];
    const float* W_out       = (const float*)d_in[7];
    const float* b_out       = (const float*)d_in[8];
    float* out = (float*)d_out;

    // workspace layout (floats): att_h [256*1024] | weight [256*256] | att_res [256*2048]
    float* att_h   = (float*)d_ws;
    float* weight  = att_h  + (size_t)BATCH * (2 * ATT_HID);
    float* att_res = weight + (size_t)BATCH * WPAD;

    // 1) att_h = h @ W_h2att + b       (WMMA f32)
    k_gemm_h2att<<<dim3(16, 16), 128, 0, stream>>>(h, W_h2att, b_h2att, att_h);
    // 2) gated scores + softmax        (streams p_att_feats, 206 MB, NT)
    k_scores_softmax<<<BATCH, 256, 0, stream>>>(p_att_feats, att_h, w_alpha, b_alpha, weight);
    // 3) att_res = weight . att_feats  (streams att_feats, 411 MB, NT + prefetch)
    k_att_res<<<BATCH, 256, 0, stream>>>(att_feats, weight, att_res);
    // 4) out GEMM + fused tanh*sigmoid gate (WMMA f32, dual-tile)
    k_gemm_out_gate<<<dim3(16, 16), 128, 0, stream>>>(att_res, W_out, b_out, out);
}